// staaGCN_46523085750314
// MI455X (gfx1250) — compile-verified
//
#include <hip/hip_runtime.h>
#include <math.h>

#define Tn 12
#define Bn 256
#define Nn 35
#define Mn 29
#define Hn 64
#define Rn 7
#define NBn 3
#define HEADSn 8

typedef __attribute__((ext_vector_type(2))) float v2f;
typedef __attribute__((ext_vector_type(8))) float v8f;

// Full-precision CDNA5 matrix op: D(16x16,f32) = A(16x4,f32) x B(4x16,f32) + C
__device__ __forceinline__ v8f wmma4(v2f a, v2f b, v8f c) {
  return __builtin_amdgcn_wmma_f32_16x16x4_f32(
      /*neg_a=*/false, a, /*neg_b=*/false, b,
      /*c_mod=*/(short)0, c, /*reuse_a=*/false, /*reuse_b=*/false);
}

// CDNA5 async global->LDS copies (ASYNCcnt path, GVS addressing:
// mem = SADDR(sgpr64) + VADDR(byte offset), LDS dst offset in VGPR).
__device__ __forceinline__ void async_ld_b32(unsigned lds_off,
                                             const float* __restrict__ gbase,
                                             unsigned byte_off) {
  asm volatile("global_load_async_to_lds_b32 %0, %1, %2"
               :: "v"(lds_off), "v"(byte_off), "s"(gbase) : "memory");
}
__device__ __forceinline__ void async_ld_b128(unsigned lds_off,
                                              const float* __restrict__ gbase,
                                              unsigned byte_off) {
  asm volatile("global_load_async_to_lds_b128 %0, %1, %2"
               :: "v"(lds_off), "v"(byte_off), "s"(gbase) : "memory");
}
__device__ __forceinline__ void wait_async0() {
  asm volatile("s_wait_asynccnt 0" ::: "memory");
}

// ---------------------------------------------------------------------------
// K0: Wrel[r,m,h] = sum_nb coef[r,nb] * V[nb,m,h]   (one-shot, tiny)
// ---------------------------------------------------------------------------
__global__ void wrel_kernel(const float* __restrict__ coef,
                            const float* __restrict__ V,
                            float* __restrict__ wrel) {
  const int MH = Mn * Hn;
  for (int i = threadIdx.x; i < Rn * MH; i += 256) {
    int r = i / MH, mh = i - r * MH;
    float acc = 0.f;
#pragma unroll
    for (int nb = 0; nb < NBn; ++nb) acc += coef[r * NBn + nb] * V[nb * MH + mh];
    wrel[i] = acc;
  }
}

// ---------------------------------------------------------------------------
// K1: per-batch attention -> S[b][35][35] (mean over 8 heads of row softmax)
// ---------------------------------------------------------------------------
__device__ __forceinline__ float dot8(const float* __restrict__ q,
                                      const float* __restrict__ k) {
  float s = 0.f;
#pragma unroll
  for (int d = 0; d < 8; ++d) s += q[d] * k[d];
  return s;
}

__global__ void attn_kernel(const float* __restrict__ xt,
                            const float* __restrict__ Wq,
                            const float* __restrict__ Wk,
                            float* __restrict__ Sout) {
  __shared__ float sX[Nn * Mn];
  __shared__ float sWq[Mn * Hn];
  __shared__ float sWk[Mn * Hn];
  __shared__ float sQ[Nn * Hn];
  __shared__ float sK[Nn * Hn];
  __shared__ float sS[Nn * Nn];
  const int tid = threadIdx.x;
  const int b = blockIdx.x;
  const float* xb = xt + (size_t)b * Nn * Mn;

  for (int i = tid; i < Nn * Mn; i += 256) sX[i] = xb[i];
  for (int i = tid; i < Mn * Hn; i += 256) { sWq[i] = Wq[i]; sWk[i] = Wk[i]; }
  for (int i = tid; i < Nn * Nn; i += 256) sS[i] = 0.f;
  __syncthreads();

  // q = x @ Wq, k = x @ Wk  (35x29 @ 29x64)
  for (int i = tid; i < Nn * Hn; i += 256) {
    int n = i >> 6, c = i & 63;
    float aq = 0.f, ak = 0.f;
    for (int m = 0; m < Mn; ++m) {
      float xv = sX[n * Mn + m];
      aq += xv * sWq[m * Hn + c];
      ak += xv * sWk[m * Hn + c];
    }
    sQ[i] = aq;
    sK[i] = ak;
  }
  __syncthreads();

  // 280 (head,row) softmax tasks; recompute dots (8 FMAs) instead of spilling
  const float scale = 0.35355339059327373f;  // 1/sqrt(8)
  for (int task = tid; task < Nn * HEADSn; task += 256) {
    int h = task & 7, n = task >> 3;
    const float* qr = &sQ[n * Hn + h * 8];
    float mx = -1e30f;
    for (int j = 0; j < Nn; ++j)
      mx = fmaxf(mx, dot8(qr, &sK[j * Hn + h * 8]) * scale);
    float sum = 0.f;
    for (int j = 0; j < Nn; ++j)
      sum += __expf(dot8(qr, &sK[j * Hn + h * 8]) * scale - mx);
    float inv = 0.125f / sum;  // mean over 8 heads folded in
    for (int j = 0; j < Nn; ++j) {
      float p = __expf(dot8(qr, &sK[j * Hn + h * 8]) * scale - mx) * inv;
      atomicAdd(&sS[n * Nn + j], p);  // ds_add_f32
    }
  }
  __syncthreads();

  float* Sb = Sout + (size_t)b * Nn * Nn;
  for (int i = tid; i < Nn * Nn; i += 256) Sb[i] = sS[i];
}

// ---------------------------------------------------------------------------
// K2: column-wise l2 normalization of As*S, Af*S, At*S and 7-relation stack
// ---------------------------------------------------------------------------
__global__ void adj_kernel(const float* __restrict__ Sin,
                           const float* __restrict__ As,
                           const float* __restrict__ Af,
                           const float* __restrict__ At,
                           float* __restrict__ Aout) {
  __shared__ float sS[Nn * Nn], sA0[Nn * Nn], sA1[Nn * Nn], sA2[Nn * Nn];
  __shared__ float nrm0[Nn], nrm1[Nn], nrm2[Nn];
  const int tid = threadIdx.x;
  const int b = blockIdx.x;
  const float* Sb = Sin + (size_t)b * Nn * Nn;
  for (int i = tid; i < Nn * Nn; i += 256) {
    sS[i] = Sb[i]; sA0[i] = As[i]; sA1[i] = Af[i]; sA2[i] = At[i];
  }
  __syncthreads();
  if (tid < Nn) {
    int j = tid;
    float a0 = 0.f, a1 = 0.f, a2 = 0.f;
    for (int n = 0; n < Nn; ++n) {
      float s = sS[n * Nn + j];
      float v0 = sA0[n * Nn + j] * s, v1 = sA1[n * Nn + j] * s, v2 = sA2[n * Nn + j] * s;
      a0 += v0 * v0; a1 += v1 * v1; a2 += v2 * v2;
    }
    nrm0[j] = fmaxf(sqrtf(a0), 1e-12f);
    nrm1[j] = fmaxf(sqrtf(a1), 1e-12f);
    nrm2[j] = fmaxf(sqrtf(a2), 1e-12f);
  }
  __syncthreads();
  const size_t bs = (size_t)Bn * Nn * Nn;
  float* ob = Aout + (size_t)b * Nn * Nn;
  for (int e = tid; e < Nn * Nn; e += 256) {
    int j = e % Nn;
    float s = sS[e];
    float a0 = sA0[e] * s / nrm0[j];
    float a1 = sA1[e] * s / nrm1[j];
    float a2 = sA2[e] * s / nrm2[j];
    ob[0 * bs + e] = a0;
    ob[1 * bs + e] = a1;
    ob[2 * bs + e] = a2;
    ob[3 * bs + e] = (a0 + a1) * 0.5f;
    ob[4 * bs + e] = (a0 + a2) * 0.5f;
    ob[5 * bs + e] = (a1 + a2) * 0.5f;
    ob[6 * bs + e] = (a0 + a1 + a2) * (1.f / 3.f);
  }
}

// ---------------------------------------------------------------------------
// K3: msg[r,b] = relu( (A_r @ x_t) @ Wrel_r )  via f32 WMMA
//   A_r: 35x35 (pad 48x36), x: 35x29 (pad 36x32), Wrel_r: 29x64
//   Inputs staged into padded LDS tiles with async global->LDS scatter.
// ---------------------------------------------------------------------------
__global__ void msg_kernel(const float* __restrict__ Astack,
                           const float* __restrict__ xt,
                           const float* __restrict__ wrel,
                           float* __restrict__ msg) {
  __shared__ __attribute__((aligned(16))) float sA[48 * 36];
  __shared__ __attribute__((aligned(16))) float sX[36 * 32];
  __shared__ __attribute__((aligned(16))) float sAX[48 * 32];
  const int tid = threadIdx.x;
  const int r = blockIdx.x / Bn;
  const int b = blockIdx.x - r * Bn;

  for (int i = tid; i < 48 * 36; i += 256) sA[i] = 0.f;
  for (int i = tid; i < 36 * 32; i += 256) sX[i] = 0.f;
  __syncthreads();  // DS zero-stores complete before async LDS writes land

  const float* Ar = Astack + ((size_t)r * Bn + b) * Nn * Nn;
  const float* xb = xt + (size_t)b * Nn * Mn;
  for (int i = tid; i < Nn * Nn; i += 256) {
    int n = i / Nn, j = i - n * Nn;
    async_ld_b32((unsigned)(size_t)&sA[n * 36 + j], Ar, (unsigned)(i * 4));
  }
  for (int i = tid; i < Nn * Mn; i += 256) {
    int n = i / Mn, m = i - n * Mn;
    async_ld_b32((unsigned)(size_t)&sX[n * 32 + m], xb, (unsigned)(i * 4));
  }
  wait_async0();
  __syncthreads();

  const int wave = tid >> 5, lane = tid & 31;
  const int row16 = lane & 15, half = lane >> 4;

  // ax = A_r @ x : 6 tiles (3 row-tiles x 2 col-tiles), K = 36
  if (wave < 6) {
    int mt = wave >> 1, ct = wave & 1;
    v8f acc = {};
    for (int kk = 0; kk < 9; ++kk) {
      int kb = kk * 4 + half * 2;
      v2f a = { sA[(mt * 16 + row16) * 36 + kb], sA[(mt * 16 + row16) * 36 + kb + 1] };
      v2f bv = { sX[kb * 32 + ct * 16 + row16], sX[(kb + 1) * 32 + ct * 16 + row16] };
      acc = wmma4(a, bv, acc);
    }
    int col = ct * 16 + row16;
    int rbase = mt * 16 + half * 8;
#pragma unroll
    for (int i = 0; i < 8; ++i) sAX[(rbase + i) * 32 + col] = acc[i];
  }
  __syncthreads();

  // msg = relu(ax @ Wrel_r) : 12 tiles, K = 32 (rows >= 29 of Wrel guarded)
  const float* Wr = wrel + (size_t)r * Mn * Hn;
  float* mg = msg + ((size_t)r * Bn + b) * Nn * Hn;
  for (int tile = wave; tile < 12; tile += 8) {
    int mt = tile >> 2, ct = tile & 3;
    int col = ct * 16 + row16;
    v8f acc = {};
    for (int kk = 0; kk < 8; ++kk) {
      int kb = kk * 4 + half * 2;
      v2f a = { sAX[(mt * 16 + row16) * 32 + kb], sAX[(mt * 16 + row16) * 32 + kb + 1] };
      v2f bv = { (kb < Mn) ? Wr[kb * Hn + col] : 0.f,
                 (kb + 1 < Mn) ? Wr[(kb + 1) * Hn + col] : 0.f };
      acc = wmma4(a, bv, acc);
    }
    int rbase = mt * 16 + half * 8;
#pragma unroll
    for (int i = 0; i < 8; ++i) {
      int row = rbase + i;
      if (row < Nn) mg[row * Hn + col] = fmaxf(acc[i], 0.f);
    }
  }
}

// ---------------------------------------------------------------------------
// K4: fused GRU update per (r,b): six [48x64]@[64x64] WMMA matmuls + gating.
// msg/h staged via async b128 global->LDS; weights prefetched (L2-hot).
// ---------------------------------------------------------------------------
__global__ void gru_kernel(const float* __restrict__ msg,
                           const float* __restrict__ hprev,
                           const float* __restrict__ Uz, const float* __restrict__ Wz, const float* __restrict__ bz,
                           const float* __restrict__ Ur, const float* __restrict__ Wrg, const float* __restrict__ brg,
                           const float* __restrict__ Uh, const float* __restrict__ Wh, const float* __restrict__ bh,
                           float* __restrict__ hout) {
  __shared__ __attribute__((aligned(16))) float sM[48 * 64];
  __shared__ __attribute__((aligned(16))) float sH[48 * 64];
  __shared__ __attribute__((aligned(16))) float sZ[48 * 64];
  __shared__ __attribute__((aligned(16))) float sR[48 * 64];
  const int tid = threadIdx.x;
  const size_t base = (size_t)blockIdx.x * Nn * Hn;

  // Async-stage msg and h_prev: 2240 floats each = 560 x 16B transfers.
  const float* gm = msg + base;
  const float* gh = hprev + base;
  for (int i = tid; i < 560; i += 256) {
    async_ld_b128((unsigned)(size_t)&sM[i * 4], gm, (unsigned)(i * 16));
    async_ld_b128((unsigned)(size_t)&sH[i * 4], gh, (unsigned)(i * 16));
  }
  // Zero pad rows (disjoint LDS range; overlaps with async copies).
  for (int i = 2240 + tid; i < 3072; i += 256) { sM[i] = 0.f; sH[i] = 0.f; }
  // Warm L2/L0 for the six weight matrices (16KB each, 256B stride covers all).
  if (tid < 64) {
    __builtin_prefetch(Uz + tid * 64, 0, 1);
    __builtin_prefetch(Wz + tid * 64, 0, 1);
    __builtin_prefetch(Ur + tid * 64, 0, 1);
    __builtin_prefetch(Wrg + tid * 64, 0, 1);
    __builtin_prefetch(Uh + tid * 64, 0, 1);
    __builtin_prefetch(Wh + tid * 64, 0, 1);
  }
  wait_async0();
  __syncthreads();

  const int wave = tid >> 5, lane = tid & 31;
  const int row16 = lane & 15, half = lane >> 4;

  // Phase 1: z and r gates -> 24 tile-tasks over 8 waves (wave-uniform)
  for (int task = wave; task < 24; task += 8) {
    int gate = task / 12;
    int tile = task - gate * 12;
    int mt = tile >> 2, ct = tile & 3;
    const float* U = gate ? Ur : Uz;
    const float* W = gate ? Wrg : Wz;
    const float* bb = gate ? brg : bz;
    float* dst = gate ? sR : sZ;
    int col = ct * 16 + row16;
    v8f acc = {};
    for (int kk = 0; kk < 16; ++kk) {
      int kb = kk * 4 + half * 2;
      v2f a = { sM[(mt * 16 + row16) * 64 + kb], sM[(mt * 16 + row16) * 64 + kb + 1] };
      v2f bv = { U[kb * 64 + col], U[(kb + 1) * 64 + col] };
      acc = wmma4(a, bv, acc);
    }
    for (int kk = 0; kk < 16; ++kk) {
      int kb = kk * 4 + half * 2;
      v2f a = { sH[(mt * 16 + row16) * 64 + kb], sH[(mt * 16 + row16) * 64 + kb + 1] };
      v2f bv = { W[kb * 64 + col], W[(kb + 1) * 64 + col] };
      acc = wmma4(a, bv, acc);
    }
    float bias = bb[col];
    int rbase = mt * 16 + half * 8;
#pragma unroll
    for (int i = 0; i < 8; ++i)
      dst[(rbase + i) * 64 + col] = 1.f / (1.f + __expf(-(acc[i] + bias)));
  }
  __syncthreads();

  // Phase 2: r <- r * h (pad rows become 0 since h pad is 0)
  for (int i = tid; i < 3072; i += 256) sR[i] *= sH[i];
  __syncthreads();

  // Phase 3: hh = tanh(msg@Uh + (r*h)@Wh + bh); h' = (1-z)*h + z*hh
  for (int tile = wave; tile < 12; tile += 8) {
    int mt = tile >> 2, ct = tile & 3;
    int col = ct * 16 + row16;
    v8f acc = {};
    for (int kk = 0; kk < 16; ++kk) {
      int kb = kk * 4 + half * 2;
      v2f a = { sM[(mt * 16 + row16) * 64 + kb], sM[(mt * 16 + row16) * 64 + kb + 1] };
      v2f bv = { Uh[kb * 64 + col], Uh[(kb + 1) * 64 + col] };
      acc = wmma4(a, bv, acc);
    }
    for (int kk = 0; kk < 16; ++kk) {
      int kb = kk * 4 + half * 2;
      v2f a = { sR[(mt * 16 + row16) * 64 + kb], sR[(mt * 16 + row16) * 64 + kb + 1] };
      v2f bv = { Wh[kb * 64 + col], Wh[(kb + 1) * 64 + col] };
      acc = wmma4(a, bv, acc);
    }
    float bias = bh[col];
    int rbase = mt * 16 + half * 8;
#pragma unroll
    for (int i = 0; i < 8; ++i) {
      int row = rbase + i;
      if (row < Nn) {
        float hh = tanhf(acc[i] + bias);
        float z = sZ[row * 64 + col];
        float hp = sH[row * 64 + col];
        hout[base + row * Hn + col] = (1.f - z) * hp + z * hh;
      }
    }
  }
}

// ---------------------------------------------------------------------------
extern "C" void kernel_launch(void* const* d_in, const int* in_sizes, int n_in,
                              void* d_out, int out_size, void* d_ws, size_t ws_size,
                              hipStream_t stream) {
  (void)in_sizes; (void)n_in; (void)out_size; (void)ws_size;
  const float* x    = (const float*)d_in[0];
  const float* As   = (const float*)d_in[1];
  const float* Af   = (const float*)d_in[2];
  const float* At   = (const float*)d_in[3];
  const float* h0   = (const float*)d_in[4];
  const float* Wq   = (const float*)d_in[5];
  const float* Wk   = (const float*)d_in[6];
  const float* V    = (const float*)d_in[7];
  const float* coef = (const float*)d_in[8];
  const float* Uz   = (const float*)d_in[9];
  const float* Wz   = (const float*)d_in[10];
  const float* bz   = (const float*)d_in[11];
  const float* Ur   = (const float*)d_in[12];
  const float* Wrg  = (const float*)d_in[13];
  const float* brg  = (const float*)d_in[14];
  const float* Uh   = (const float*)d_in[15];
  const float* Wh   = (const float*)d_in[16];
  const float* bh   = (const float*)d_in[17];
  float* out = (float*)d_out;
  float* ws  = (float*)d_ws;

  // workspace layout (floats)
  float* wrel = ws;                  // 7*29*64   = 12992 (rounded to 13056)
  float* Sbuf = ws + 13056;          // 256*1225  = 313600
  float* Abuf = Sbuf + 313600;       // 7*256*1225 = 2195200
  float* msgb = Abuf + 2195200;      // 7*256*2240 = 4014080  (total ~26 MB)

  wrel_kernel<<<1, 256, 0, stream>>>(coef, V, wrel);

  const size_t stepOut = (size_t)Rn * Bn * Nn * Hn;
  for (int t = 0; t < Tn; ++t) {
    const float* xt = x + (size_t)t * Bn * Nn * Mn;
    const float* hp = (t == 0) ? h0 : (out + (size_t)(t - 1) * stepOut);
    float* ho = out + (size_t)t * stepOut;
    attn_kernel<<<Bn, 256, 0, stream>>>(xt, Wq, Wk, Sbuf);
    adj_kernel<<<Bn, 256, 0, stream>>>(Sbuf, As, Af, At, Abuf);
    msg_kernel<<<Rn * Bn, 256, 0, stream>>>(Abuf, xt, wrel, msgb);
    gru_kernel<<<Rn * Bn, 256, 0, stream>>>(msgb, hp,
                                            Uz, Wz, bz, Ur, Wrg, brg, Uh, Wh, bh, ho);
  }
}